// LowFreqCrossAttn_46420006535708
// MI455X (gfx1250) — compile-verified
//
#include <hip/hip_runtime.h>
#include <hip/hip_bf16.h>

typedef __attribute__((ext_vector_type(2))) float v2f;
typedef __attribute__((ext_vector_type(4))) float v4f;
typedef __attribute__((ext_vector_type(8))) float v8f;

#define RES 28
#define DIMC 384
#define NHEAD 8
#define HDIM 48
#define NTOK 784   // RES*RES
#define BATCH 16

// LDS strides (floats)
#define QS_S 136   // [d][token], 2*136 % 64banks == 16 -> halves disjoint
#define KS_S 40    // [d][key],   2*40  % 64 == 16
#define VS_S 36    // [d'][key]-> stored [d][?]; strided-lane reads: 36 conflict-free
#define PS_S 33
#define AS_S 36
#define BS_S 72

__device__ __forceinline__ float hmax16(float v) {
#pragma unroll
    for (int m = 8; m >= 1; m >>= 1) v = fmaxf(v, __shfl_xor(v, m, 32));
    return v;
}
__device__ __forceinline__ float hsum16(float v) {
#pragma unroll
    for (int m = 8; m >= 1; m >>= 1) v += __shfl_xor(v, m, 32);
    return v;
}

// Expand attn_biases[h, bias_idxs[n,m]] -> biasf[h, n, m]  (8 x 784 x 784)
__global__ void bias_expand_kernel(const float* __restrict__ biases,
                                   const int* __restrict__ idxs,
                                   float* __restrict__ outb, int noff) {
    int i = blockIdx.x * 256 + threadIdx.x;
    if (i >= NTOK * NTOK) return;
    int id = idxs[i];
#pragma unroll
    for (int h = 0; h < NHEAD; ++h)
        outb[(size_t)h * NTOK * NTOK + i] = biases[h * noff + id];
}

// 1x1-conv GEMM: Y[b, o, n] = sum_c W[o,c] * X[b, c, n] + bias[o]
// Block: 256 threads = 8 waves, tile M=32 x N=64, each wave one 16x16 WMMA tile.
__global__ __launch_bounds__(256) void gemm1x1_kernel(
    const float* __restrict__ W,     // O x C row-major
    const float* __restrict__ X,     // batches of C x Ncols
    const float* __restrict__ bias,  // O
    float* __restrict__ Y,           // batches of O x Ncols
    int O, int C, int Ncols) {
    __shared__ float As[32 * AS_S];
    __shared__ float Bs[32 * BS_S];
    const int tid = threadIdx.x;
    const int wave = tid >> 5;
    const int lane = tid & 31;
    const int half = lane >> 4;
    const int l = lane & 15;
    const int mo = (wave >> 2) * 16;   // 0 or 16
    const int no = (wave & 3) * 16;    // 0..48
    const int o0 = blockIdx.y * 32;
    const int n0 = blockIdx.x * 64;
    const size_t batch = blockIdx.z;
    const float* Xb = X + batch * (size_t)C * Ncols;
    float* Yb = Y + batch * (size_t)O * Ncols;

    // staging assignments: one v4f of A, two v4f of B per thread per K-chunk
    const int ar = tid >> 3, ac4 = (tid & 7) * 4;    // A: 32 rows x 8 vec4
    const int br = tid >> 4, bc4 = (tid & 15) * 4;   // B: 16 rows x 16 vec4 (x2)
    const bool bvalid = (n0 + bc4) < Ncols;          // NTOK%4==0 -> full or empty

    v8f acc = {};
    for (int k0 = 0; k0 < C; k0 += 32) {
        __syncthreads();
        v4f av = *(const v4f*)&W[(size_t)(o0 + ar) * C + k0 + ac4];
        v4f bv0 = {}, bv1 = {};
        if (bvalid) {
            bv0 = *(const v4f*)&Xb[(size_t)(k0 + br) * Ncols + n0 + bc4];
            bv1 = *(const v4f*)&Xb[(size_t)(k0 + br + 16) * Ncols + n0 + bc4];
        }
        *(v4f*)&As[ar * AS_S + ac4] = av;
        *(v4f*)&Bs[br * BS_S + bc4] = bv0;
        *(v4f*)&Bs[(br + 16) * BS_S + bc4] = bv1;
        if (k0 + 32 < C) {
            __builtin_prefetch(&W[(size_t)(o0 + ar) * C + k0 + 32 + ac4], 0, 3);
            if (bvalid)
                __builtin_prefetch(&Xb[(size_t)(k0 + 32 + br) * Ncols + n0 + bc4], 0, 3);
        }
        __syncthreads();
#pragma unroll
        for (int kk = 0; kk < 32; kk += 4) {
            int ka = kk + 2 * half;
            // A frag: lanes 0-15 -> K=kk,kk+1 ; lanes 16-31 -> K=kk+2,kk+3 (M=l)
            v2f a = { As[(mo + l) * AS_S + ka], As[(mo + l) * AS_S + ka + 1] };
            v2f b = { Bs[ka * BS_S + no + l], Bs[(ka + 1) * BS_S + no + l] };
            acc = __builtin_amdgcn_wmma_f32_16x16x4_f32(false, a, false, b,
                                                        (short)0, acc, false, false);
        }
    }
#pragma unroll
    for (int r = 0; r < 8; ++r) {
        int o = o0 + mo + r + 8 * half;   // D frag: VGPR r = row r / r+8
        int n = n0 + no + l;
        if (n < Ncols) Yb[(size_t)o * Ncols + n] = acc[r] + bias[o];
    }
}

// Fused flash-attention per (b, h). Block = 8 waves; 128 query rows per block,
// each wave owns 16 query rows; loop key tiles of 32 with online softmax.
__global__ __launch_bounds__(256) void attn_fused_kernel(
    const float* __restrict__ q,      // B x DIM x N  (d-major, token minor)
    const float* __restrict__ kv,     // B x 2DIM x N (k rows then v rows)
    const float* __restrict__ biasf,  // NH x N x N
    float* __restrict__ out) {        // B x DIM x N
    __shared__ float Qs[HDIM * QS_S];     // [d][query token]
    __shared__ float Ks[HDIM * KS_S];     // [d][key]
    __shared__ float Vs[HDIM * VS_S];     // [d][key]
    __shared__ float Ps[8 * 16 * PS_S];   // per-wave P tile [row][key]

    const int tid = threadIdx.x;
    const int wave = tid >> 5;
    const int lane = tid & 31;
    const int half = lane >> 4;
    const int l = lane & 15;
    const int h = blockIdx.y, b = blockIdx.z;
    const int qbase = blockIdx.x * 128;
    const int q0_raw = qbase + wave * 16;
    const bool wvalid = (q0_raw < NTOK);

    const size_t qhead = ((size_t)b * DIMC + h * HDIM) * NTOK;
    const size_t khead = ((size_t)b * 2 * DIMC + h * HDIM) * NTOK;
    const size_t vhead = ((size_t)b * 2 * DIMC + DIMC + h * HDIM) * NTOK;
    float* Psw = &Ps[wave * 16 * PS_S];

    // stage Q: Qs[d][i] = q[d, qbase+i] (tokens clamped for tail block)
    for (int idx = tid; idx < (HDIM * 128) / 4; idx += 256) {
        int i4 = (idx & 31) * 4;
        int d = idx >> 5;
        int ng = qbase + i4;
        v4f qv;
        if (ng < NTOK) {                      // NTOK%4==0 -> full vec valid
            qv = *(const v4f*)&q[qhead + (size_t)d * NTOK + ng];
        } else {
            float s = q[qhead + (size_t)d * NTOK + (NTOK - 1)];
            qv = (v4f){s, s, s, s};
        }
        *(v4f*)&Qs[d * QS_S + i4] = qv;
    }

    v8f acc0 = {}, acc1 = {}, acc2 = {};
    float rowmax[8], rowsum[8];
#pragma unroll
    for (int r = 0; r < 8; ++r) { rowmax[r] = -1e30f; rowsum[r] = 0.0f; }

    const float scale = 0.14433756729740643f;  // 48^-0.5
    const int qrow = wave * 16 + l;
    const int sd = tid >> 3, sm4 = (tid & 7) * 4;  // K/V staging map (384 vec4)

    for (int t = 0; t < 25; ++t) {
        const int m0 = t * 32;
        __syncthreads();  // prev-iter K/V reads done (covers initial Q stage too)
        if (m0 + 32 <= NTOK) {
            // fully in-bounds tile: async DMA global->LDS, 16B per lane
            for (int idx = tid; idx < (HDIM * 32) / 4; idx += 256) {
                int m4 = (idx & 7) * 4;
                int d = idx >> 3;
                const float* gk = &kv[khead + (size_t)d * NTOK + m0 + m4];
                const float* gv = &kv[vhead + (size_t)d * NTOK + m0 + m4];
                unsigned lk = (unsigned)(size_t)&Ks[d * KS_S + m4];
                unsigned lv = (unsigned)(size_t)&Vs[d * VS_S + m4];
                asm volatile("global_load_async_to_lds_b128 %0, %1, off"
                             :: "v"(lk), "v"(gk) : "memory");
                asm volatile("global_load_async_to_lds_b128 %0, %1, off"
                             :: "v"(lv), "v"(gv) : "memory");
            }
            if (m0 + 64 <= NTOK) {
                __builtin_prefetch(&kv[khead + (size_t)sd * NTOK + m0 + 32 + sm4], 0, 3);
                __builtin_prefetch(&kv[vhead + (size_t)sd * NTOK + m0 + 32 + sm4], 0, 3);
            }
            asm volatile("s_wait_asynccnt 0" ::: "memory");
        } else {
            // tail tile: guarded scalar staging with zero fill
            for (int idx = tid; idx < HDIM * 32; idx += 256) {
                int m = idx & 31, d = idx >> 5;
                int mg = m0 + m;
                float kvl = (mg < NTOK) ? kv[khead + (size_t)d * NTOK + mg] : 0.0f;
                float vvl = (mg < NTOK) ? kv[vhead + (size_t)d * NTOK + mg] : 0.0f;
                Ks[d * KS_S + m] = kvl;
                Vs[d * VS_S + m] = vvl;
            }
        }
        __syncthreads();

        // S = Q^T K : two 16x16 subtiles over K-dim HD=48
        v8f s0 = {}, s1 = {};
#pragma unroll
        for (int d0 = 0; d0 < HDIM; d0 += 4) {
            int ka = d0 + 2 * half;
            v2f a = { Qs[ka * QS_S + qrow], Qs[(ka + 1) * QS_S + qrow] };
            v2f b0 = { Ks[ka * KS_S + l], Ks[(ka + 1) * KS_S + l] };
            s0 = __builtin_amdgcn_wmma_f32_16x16x4_f32(false, a, false, b0,
                                                       (short)0, s0, false, false);
            v2f b1 = { Ks[ka * KS_S + 16 + l], Ks[(ka + 1) * KS_S + 16 + l] };
            s1 = __builtin_amdgcn_wmma_f32_16x16x4_f32(false, a, false, b1,
                                                       (short)0, s1, false, false);
        }

        // online softmax update (lanes<16: row r, lanes>=16: row r+8)
#pragma unroll
        for (int r = 0; r < 8; ++r) {
            int ngrow = q0_raw + r + 8 * half; if (ngrow > NTOK - 1) ngrow = NTOK - 1;
            const float* brow = biasf + ((size_t)h * NTOK + ngrow) * NTOK;
            int mg0 = m0 + l, mg1 = m0 + 16 + l;
            float v0 = (mg0 < NTOK) ? s0[r] * scale + brow[mg0] : -1e30f;
            float v1 = (mg1 < NTOK) ? s1[r] * scale + brow[mg1] : -1e30f;
            float mnew = fmaxf(rowmax[r], fmaxf(hmax16(v0), hmax16(v1)));
            float corr = __expf(rowmax[r] - mnew);
            rowmax[r] = mnew;
            float p0 = __expf(v0 - mnew);
            float p1 = __expf(v1 - mnew);
            rowsum[r] = rowsum[r] * corr + hsum16(p0) + hsum16(p1);
            Psw[(r + 8 * half) * PS_S + l] = p0;
            Psw[(r + 8 * half) * PS_S + 16 + l] = p1;
            acc0[r] *= corr; acc1[r] *= corr; acc2[r] *= corr;
        }

        // O += P (16x32) x V (32x48); intra-wave LDS RAW is in-order per ISA
#pragma unroll
        for (int k0 = 0; k0 < 32; k0 += 4) {
            int ka = k0 + 2 * half;
            v2f a = { Psw[l * PS_S + ka], Psw[l * PS_S + ka + 1] };
            v2f b0 = { Vs[(l) * VS_S + ka],      Vs[(l) * VS_S + ka + 1] };
            v2f b1 = { Vs[(16 + l) * VS_S + ka], Vs[(16 + l) * VS_S + ka + 1] };
            v2f b2 = { Vs[(32 + l) * VS_S + ka], Vs[(32 + l) * VS_S + ka + 1] };
            acc0 = __builtin_amdgcn_wmma_f32_16x16x4_f32(false, a, false, b0,
                                                         (short)0, acc0, false, false);
            acc1 = __builtin_amdgcn_wmma_f32_16x16x4_f32(false, a, false, b1,
                                                         (short)0, acc1, false, false);
            acc2 = __builtin_amdgcn_wmma_f32_16x16x4_f32(false, a, false, b2,
                                                         (short)0, acc2, false, false);
        }
    }

    if (wvalid) {
#pragma unroll
        for (int r = 0; r < 8; ++r) {
            float inv = 1.0f / rowsum[r];
            int ng = q0_raw + r + 8 * half;
            out[qhead + (size_t)(l) * NTOK + ng]      = acc0[r] * inv;
            out[qhead + (size_t)(16 + l) * NTOK + ng] = acc1[r] * inv;
            out[qhead + (size_t)(32 + l) * NTOK + ng] = acc2[r] * inv;
        }
    }
}

extern "C" void kernel_launch(void* const* d_in, const int* in_sizes, int n_in,
                              void* d_out, int out_size, void* d_ws, size_t ws_size,
                              hipStream_t stream) {
    (void)n_in; (void)out_size; (void)ws_size;
    const float* ll   = (const float*)d_in[0];
    const float* ha   = (const float*)d_in[1];
    const float* q_w  = (const float*)d_in[2];
    const float* q_b  = (const float*)d_in[3];
    const float* kv_w = (const float*)d_in[4];
    const float* kv_b = (const float*)d_in[5];
    const float* p_w  = (const float*)d_in[6];
    const float* p_b  = (const float*)d_in[7];
    const float* ab   = (const float*)d_in[8];
    const int*   idxs = (const int*)d_in[9];
    float* out = (float*)d_out;
    const int noff = in_sizes[8] / NHEAD;

    float* q_buf  = (float*)d_ws;                              // B*DIM*N
    float* kv_buf = q_buf  + (size_t)BATCH * DIMC * NTOK;      // B*2DIM*N
    float* attn_o = kv_buf + (size_t)BATCH * 2 * DIMC * NTOK;  // B*DIM*N
    float* biasf  = attn_o + (size_t)BATCH * DIMC * NTOK;      // NH*N*N

    bias_expand_kernel<<<(NTOK * NTOK + 255) / 256, 256, 0, stream>>>(ab, idxs, biasf, noff);

    gemm1x1_kernel<<<dim3(13, DIMC / 32, BATCH), 256, 0, stream>>>(
        q_w, ll, q_b, q_buf, DIMC, DIMC, NTOK);
    gemm1x1_kernel<<<dim3(13, 2 * DIMC / 32, BATCH), 256, 0, stream>>>(
        kv_w, ha, kv_b, kv_buf, 2 * DIMC, DIMC, NTOK);

    attn_fused_kernel<<<dim3(7, NHEAD, BATCH), 256, 0, stream>>>(
        q_buf, kv_buf, biasf, attn_o);

    gemm1x1_kernel<<<dim3(13, DIMC / 32, BATCH), 256, 0, stream>>>(
        p_w, attn_o, p_b, out, DIMC, DIMC, NTOK);
}